// Unitary_46024869544015
// MI455X (gfx1250) — compile-verified
//
#include <hip/hip_runtime.h>

typedef float v2f __attribute__((ext_vector_type(2)));
typedef float v8f __attribute__((ext_vector_type(8)));
typedef unsigned int v4u __attribute__((ext_vector_type(4)));
typedef int v4i __attribute__((ext_vector_type(4)));
typedef int v8i __attribute__((ext_vector_type(8)));

// ---------------------------------------------------------------------------
// Lane shift helper (wave32): value of `srcLane` (wraps mod 32; wrapped reads
// are multiplied by zero coefficients so garbage is harmless).
// ---------------------------------------------------------------------------
__device__ __forceinline__ float lane_read(float v, int srcLane) {
    int iv = __float_as_int(v);
    int r  = __builtin_amdgcn_ds_bpermute(srcLane << 2, iv);
    return __int_as_float(r);
}

// ---------------------------------------------------------------------------
// Kernel 0: per-layer mixer coefficients -> table T[128][2048]
//           layout per layer: ur[512] | ui[512] | vr[512] | vi[512]
// Grid: 128 blocks (one per layer) x 256 threads (one per pair).
// ---------------------------------------------------------------------------
__global__ __launch_bounds__(256) void coef_table_kernel(
    const float* __restrict__ thA, const float* __restrict__ phA,
    const float* __restrict__ thB, const float* __restrict__ phB,
    float* __restrict__ T)
{
    const int layer = blockIdx.x;
    const int j     = threadIdx.x;
    const int row   = layer >> 1;
    float* ur = T + (size_t)layer * 2048;
    float* ui = ur + 512;
    float* vr = ur + 1024;
    float* vi = ur + 1536;

    if ((layer & 1) == 0) {
        // Stage A: pairs (2j, 2j+1), j = 0..255
        float th = thA[row * 256 + j];
        float ph = phA[row * 256 + j];
        float s1 = sinf(0.5f * th), c1 = cosf(0.5f * th);
        float s2 = sinf(0.5f * th + ph), c2 = cosf(0.5f * th + ph);
        int e1 = 2 * j, e2 = e1 + 1;
        ur[e1] = -s1 * s2; ui[e1] =  s1 * c2; vr[e1] = -c1 * s1; vi[e1] = c1 * c1;
        ur[e2] =  s1 * s1; ui[e2] = -c1 * s1; vr[e2] = -c1 * s2; vi[e2] = c1 * c2;
    } else {
        // Stage B: pairs (2j+1, 2j+2), j = 0..254 ; elements 0,511 identity
        if (j < 255) {
            float th = thB[row * 255 + j];
            float ph = phB[row * 255 + j];
            float s1 = sinf(0.5f * th), c1 = cosf(0.5f * th);
            float s2 = sinf(0.5f * th + ph), c2 = cosf(0.5f * th + ph);
            int e1 = 2 * j + 1, e2 = e1 + 1;
            ur[e1] = -s1 * s2; ui[e1] =  s1 * c2; vr[e1] = -c1 * s1; vi[e1] = c1 * c1;
            ur[e2] =  s1 * s1; ui[e2] = -c1 * s1; vr[e2] = -c1 * s2; vi[e2] = c1 * c2;
        } else {
            ur[0]   = 1.0f; ui[0]   = 0.0f; vr[0]   = 0.0f; vi[0]   = 0.0f;
            ur[511] = 1.0f; ui[511] = 0.0f; vr[511] = 0.0f; vi[511] = 0.0f;
        }
    }
}

// ---------------------------------------------------------------------------
// Kernel 1: propagate 512 basis columns through 128 layers to build
// G (1024x1024) = [[Wr^T, Wi^T], [-Wi^T, Wr^T]].
// Block: 128 threads = 4 waves, one column per wave, 16 elems per lane.
// Grid: 128 blocks (128*4 = 512 columns).
// Per-layer coefficients staged into LDS: TDM prefetch (wave 0) + cooperative
// vector loads (the latter guarantee the LDS contents regardless of TDM
// descriptor semantics).
// ---------------------------------------------------------------------------
__global__ __launch_bounds__(128) void build_G_kernel(
    const float* __restrict__ T, const float* __restrict__ psi,
    float* __restrict__ G)
{
    __shared__ __align__(16) float s_coef[2048];  // ur | ui | vr | vi

    const int tid  = threadIdx.x;
    const int lane = tid & 31;
    const int wave = tid >> 5;                 // 0..3
    const int d    = blockIdx.x * 4 + wave;    // column index 0..511
    const int base = lane * 16;                // first element owned by lane

    float xr[16], xi[16];
#pragma unroll
    for (int i = 0; i < 16; ++i) {
        xr[i] = (base + i == d) ? 1.0f : 0.0f;
        xi[i] = 0.0f;
    }

    for (int layer = 0; layer < 128; ++layer) {
        const float* layT = T + (size_t)layer * 2048;

        // --- TDM async stage of the 8KB coefficient block (wave 0 only) ---
        if (tid < 32) {
            unsigned long long ga = (unsigned long long)(uintptr_t)layT;
            unsigned int lds_off  = (unsigned int)(uintptr_t)s_coef;
            v4u g0;
            g0[0] = 1u;                                   // count=1, user mode
            g0[1] = lds_off;                              // lds_addr
            g0[2] = (unsigned int)(ga & 0xffffffffu);     // global_addr[31:0]
            g0[3] = (unsigned int)((ga >> 32) & 0x1ffffffu) | (2u << 30); // [56:32] | type=2
            v8i g1;
            g1[0] = (int)(2u << 16);          // data_size=4B; no mask/pad/iter
            g1[1] = (int)(2048u << 16);       // tensor_dim0[15:0] in bits[63:48]
            g1[2] = (int)(1u << 16);          // tensor_dim0[31:16]=0 | tensor_dim1=1
            g1[3] = (int)(2048u << 16);       // tensor_dim1[31:16]=0 | tile_dim0=2048
            g1[4] = 1;                        // tile_dim1=1, tile_dim2=0
            g1[5] = 2048;                     // tensor_dim0_stride[31:0]
            g1[6] = 0;
            g1[7] = 0;
            v4i g2 = {0, 0, 0, 0};
            v4i g3 = {0, 0, 0, 0};
            v8i g4 = {0, 0, 0, 0, 0, 0, 0, 0};
            __builtin_amdgcn_tensor_load_to_lds(g0, g1, g2, g3, g4, 0);
            __builtin_amdgcn_s_wait_tensorcnt(0);
        }
        __syncthreads();

        // --- cooperative load of the same block (authoritative contents) ---
        {
            const float4* src = (const float4*)layT;
            float4* dst = (float4*)s_coef;
#pragma unroll
            for (int q = 0; q < 4; ++q) dst[tid + 128 * q] = src[tid + 128 * q];
        }
        __syncthreads();

        // --- pull this lane's coefficients into registers ---
        float Ur[16], Ui[16], Vr[16], Vi[16];
#pragma unroll
        for (int q = 0; q < 4; ++q) {
            float4 a = *(const float4*)(s_coef + base + 4 * q);
            float4 b = *(const float4*)(s_coef + 512 + base + 4 * q);
            float4 c = *(const float4*)(s_coef + 1024 + base + 4 * q);
            float4 e = *(const float4*)(s_coef + 1536 + base + 4 * q);
            Ur[4*q+0] = a.x; Ur[4*q+1] = a.y; Ur[4*q+2] = a.z; Ur[4*q+3] = a.w;
            Ui[4*q+0] = b.x; Ui[4*q+1] = b.y; Ui[4*q+2] = b.z; Ui[4*q+3] = b.w;
            Vr[4*q+0] = c.x; Vr[4*q+1] = c.y; Vr[4*q+2] = c.z; Vr[4*q+3] = c.w;
            Vi[4*q+0] = e.x; Vi[4*q+1] = e.y; Vi[4*q+2] = e.z; Vi[4*q+3] = e.w;
        }

        auto pairApply = [&](int ia, int ib) {
            float ar = xr[ia], ai = xi[ia], br = xr[ib], bi = xi[ib];
            float yar = Ur[ia]*ar - Ui[ia]*ai + Vr[ia]*br - Vi[ia]*bi;
            float yai = Ur[ia]*ai + Ui[ia]*ar + Vr[ia]*bi + Vi[ia]*br;
            float ybr = Ur[ib]*br - Ui[ib]*bi + Vr[ib]*ar - Vi[ib]*ai;
            float ybi = Ur[ib]*bi + Ui[ib]*br + Vr[ib]*ai + Vi[ib]*ar;
            xr[ia] = yar; xi[ia] = yai; xr[ib] = ybr; xi[ib] = ybi;
        };

        if ((layer & 1) == 0) {
            // Stage A: all pairs lane-internal.
#pragma unroll
            for (int o = 0; o < 16; o += 2) pairApply(o, o + 1);
        } else {
            // Stage B: fetch neighbors (pre-update values) first.
            float nr = lane_read(xr[0],  lane + 1);
            float ni = lane_read(xi[0],  lane + 1);
            float pr = lane_read(xr[15], lane - 1);
            float pi = lane_read(xi[15], lane - 1);
            float y0r = Ur[0]*xr[0]   - Ui[0]*xi[0]   + Vr[0]*pr  - Vi[0]*pi;
            float y0i = Ur[0]*xi[0]   + Ui[0]*xr[0]   + Vr[0]*pi  + Vi[0]*pr;
            float yfr = Ur[15]*xr[15] - Ui[15]*xi[15] + Vr[15]*nr - Vi[15]*ni;
            float yfi = Ur[15]*xi[15] + Ui[15]*xr[15] + Vr[15]*ni + Vi[15]*nr;
#pragma unroll
            for (int o = 1; o < 15; o += 2) pairApply(o, o + 1);
            xr[0]  = y0r; xi[0]  = y0i;
            xr[15] = yfr; xi[15] = yfi;
        }
        __syncthreads();
    }

    // Final diagonal phase e^{i psi}.
#pragma unroll
    for (int i = 0; i < 16; ++i) {
        float p = psi[base + i];
        float c = cosf(p), s = sinf(p);
        float yr = c * xr[i] - s * xi[i];
        float yi = c * xi[i] + s * xr[i];
        xr[i] = yr; xi[i] = yi;
    }

    // Rows d and d+512 of G.
    float* g0 = G + (size_t)d * 1024 + base;
    float* g1 = G + (size_t)d * 1024 + 512 + base;
    float* g2 = G + (size_t)(d + 512) * 1024 + base;
    float* g3 = G + (size_t)(d + 512) * 1024 + 512 + base;
#pragma unroll
    for (int i = 0; i < 16; ++i) {
        g0[i] = xr[i];
        g1[i] = xi[i];
        g2[i] = -xi[i];
        g3[i] = xr[i];
    }
}

// ---------------------------------------------------------------------------
// Kernel 2: C (Mx1024) = A (Mx1024) * G (1024x1024), fp32 WMMA 16x16x4.
// Block 256 threads = 8 waves in a 4(M) x 2(N) grid; each wave owns a 32x32
// output tile (4 accumulators). Block tile 128x64, BK=16.
// ---------------------------------------------------------------------------
#define BM 128
#define BN 64
#define BK 16

__global__ __launch_bounds__(256) void gemm_wmma_f32(
    const float* __restrict__ A, const float* __restrict__ B,
    float* __restrict__ C, int K, int N)
{
    __shared__ __align__(16) float As[BM * BK];  // [m][k]
    __shared__ __align__(16) float Bs[BK * BN];  // [k][n]

    const int tid  = threadIdx.x;
    const int lane = tid & 31;
    const int wave = tid >> 5;
    const int wm   = wave & 3;      // 0..3 -> row offset 32*wm
    const int wn   = wave >> 2;     // 0..1 -> col offset 32*wn
    const int m0   = blockIdx.y * BM;
    const int n0   = blockIdx.x * BN;

    const int lm = lane & 15;       // M (A-frag) / N (B,C-frag)
    const int kh = lane >> 4;       // lane-half: K offset 2*kh (A/B), M offset 8*kh (C/D)

    v8f acc00 = {}, acc01 = {}, acc10 = {}, acc11 = {};

    const int arow = tid >> 2;             // 0..63 (plus +64 row)
    const int acol = (tid & 3) * 4;        // 0..12
    const int brow = tid >> 4;             // 0..15
    const int bcol = (tid & 15) * 4;       // 0..60

    for (int k0 = 0; k0 < K; k0 += BK) {
        float4 a4a = *(const float4*)(A + (size_t)(m0 + arow) * K + k0 + acol);
        float4 a4b = *(const float4*)(A + (size_t)(m0 + arow + 64) * K + k0 + acol);
        float4 b4  = *(const float4*)(B + (size_t)(k0 + brow) * N + n0 + bcol);
        *(float4*)(As + arow * BK + acol)        = a4a;
        *(float4*)(As + (arow + 64) * BK + acol) = a4b;
        *(float4*)(Bs + brow * BN + bcol)        = b4;
        __syncthreads();

#pragma unroll
        for (int kk = 0; kk < BK; kk += 4) {
            // A fragments (16x4): lane lm = row, K = kk + 2*kh + {0,1}
            const float* ap0 = As + (32 * wm + lm) * BK + kk + 2 * kh;
            const float* ap1 = ap0 + 16 * BK;
            v2f a0; a0.x = ap0[0]; a0.y = ap0[1];
            v2f a1; a1.x = ap1[0]; a1.y = ap1[1];
            // B fragments (4x16): lane lm = col, K = kk + 2*kh + {0,1}
            const float* bp = Bs + (kk + 2 * kh) * BN + 32 * wn + lm;
            v2f b0; b0.x = bp[0];  b0.y = bp[BN];
            v2f b1; b1.x = bp[16]; b1.y = bp[BN + 16];

            acc00 = __builtin_amdgcn_wmma_f32_16x16x4_f32(false, a0, false, b0, (short)0, acc00, false, false);
            acc01 = __builtin_amdgcn_wmma_f32_16x16x4_f32(false, a0, false, b1, (short)0, acc01, false, false);
            acc10 = __builtin_amdgcn_wmma_f32_16x16x4_f32(false, a1, false, b0, (short)0, acc10, false, false);
            acc11 = __builtin_amdgcn_wmma_f32_16x16x4_f32(false, a1, false, b1, (short)0, acc11, false, false);
        }
        __syncthreads();
    }

    // C/D layout: v8f element r -> row (+ r + 8*kh), col = lm
    const int cm = m0 + 32 * wm + 8 * kh;
    const int cn = n0 + 32 * wn + lm;
#pragma unroll
    for (int r = 0; r < 8; ++r) {
        C[(size_t)(cm + r) * N + cn]            = acc00[r];
        C[(size_t)(cm + r) * N + cn + 16]       = acc01[r];
        C[(size_t)(cm + 16 + r) * N + cn]       = acc10[r];
        C[(size_t)(cm + 16 + r) * N + cn + 16]  = acc11[r];
    }
}

// ---------------------------------------------------------------------------
// Launch. Workspace layout: G (1024*1024 f32, 4MB) | T (128*2048 f32, 1MB).
// ---------------------------------------------------------------------------
extern "C" void kernel_launch(void* const* d_in, const int* in_sizes, int n_in,
                              void* d_out, int out_size, void* d_ws, size_t ws_size,
                              hipStream_t stream) {
    const float* X   = (const float*)d_in[0];
    const float* thA = (const float*)d_in[1];
    const float* phA = (const float*)d_in[2];
    const float* thB = (const float*)d_in[3];
    const float* phB = (const float*)d_in[4];
    const float* psi = (const float*)d_in[5];
    float* out = (float*)d_out;
    float* G   = (float*)d_ws;
    float* T   = G + 1024 * 1024;

    // Phase 0: per-layer 2x2 mixer coefficients.
    coef_table_kernel<<<128, 256, 0, stream>>>(thA, phA, thB, phB, T);

    // Phase 1: fold the 128-layer circuit into matrix G.
    build_G_kernel<<<128, 128, 0, stream>>>(T, psi, G);

    // Phase 2: Y = X * G via fp32 WMMA.
    const int Kdim = 1024;
    const int Ndim = 1024;
    const int Mdim = in_sizes[0] / Kdim;   // 2048
    dim3 grid(Ndim / BN, Mdim / BM);       // (16, 16)
    gemm_wmma_f32<<<grid, 256, 0, stream>>>(X, G, out, Kdim, Ndim);
}